// Winograd2d_43052752175795
// MI455X (gfx1250) — compile-verified
//
#include <hip/hip_runtime.h>

typedef __attribute__((ext_vector_type(2))) float v2f;
typedef __attribute__((ext_vector_type(8))) float v8f;

#define CIN        64
#define COUT       64
#define TQ         16                 // tiles along q per workgroup
#define CHUNK      32                 // Cin channels per LDS pass
#define XS_COLS    66                 // 16 tiles * 4 + 2 halo
#define XS_STRIDE  33                 // ch stride (coprime with 64 banks)
#define V_TSTRIDE  34                 // tile stride in V (coprime-ish, conflict-free)
#define V_ABSTRIDE (16 * V_TSTRIDE)   // 544
#define VOFF       (6 * XS_COLS * XS_STRIDE)   // 13068 floats
#define M_COSTRIDE 66
#define LDS_FLOATS (36 * 16 * M_COSTRIDE)      // 38016 floats = 152064 B
#define DYN_LDS_BYTES (LDS_FLOATS * 4)

// ---------------------------------------------------------------------------
// Repack weight (Cout, Cin, 6, 6) -> Wt[ab][ci][co] so GEMM B-fragment reads
// are contiguous across lanes (co innermost). 576 KB, stays hot in 192MB L2.
// ---------------------------------------------------------------------------
__global__ void repack_weights(const float* __restrict__ w, float* __restrict__ wt) {
    int i = blockIdx.x * 256 + threadIdx.x;          // over 36*64*64
    if (i >= 36 * CIN * COUT) return;
    int co = i & 63;
    int ci = (i >> 6) & 63;
    int ab = i >> 12;
    wt[i] = w[(co * CIN + ci) * 36 + ab];
}

// ---------------------------------------------------------------------------
// Fused Winograd: input transform -> 36 WMMA GEMMs (K=Cin) -> output transform
// One workgroup: (n, tile-row p, 16 q-tiles) x all Cout. 8 waves, wave w owns
// ab components {w, w+8, ...}.
// ---------------------------------------------------------------------------
__global__ __launch_bounds__(256)
void winograd_fused(const float* __restrict__ x,
                    const float* __restrict__ wt,
                    const float* __restrict__ bias,
                    float* __restrict__ y)
{
    extern __shared__ float smem[];

    const int tid  = threadIdx.x;
    const int lane = tid & 31;
    const int wv   = __builtin_amdgcn_readfirstlane(tid >> 5);  // wave id (scalar)
    const int half = lane >> 4;          // 0: lanes 0-15, 1: lanes 16-31
    const int l16  = lane & 15;

    const int qStart = blockIdx.x * TQ;  // tile index base along W
    const int p      = blockIdx.y;       // tile row
    const int n      = blockIdx.z;

    const v8f vzero = {0.f, 0.f, 0.f, 0.f, 0.f, 0.f, 0.f, 0.f};
    v8f acc[5][4];
#pragma unroll
    for (int j = 0; j < 5; ++j)
#pragma unroll
        for (int c = 0; c < 4; ++c)
            acc[j][c] = vzero;

    const int h0 = p * 4 - 1;
    const int w0 = qStart * 4 - 1;

    for (int chunk = 0; chunk < 2; ++chunk) {
        const int ci0 = chunk * CHUNK;

        // ---- phase 1: x -> LDS patch xs[6][66][32], zero-padded borders ----
        for (int e = tid; e < 6 * CHUNK * XS_COLS; e += 256) {
            const int c  = e % XS_COLS;           // lanes stride over c: coalesced
            const int t  = e / XS_COLS;
            const int ch = t & (CHUNK - 1);
            const int i  = t >> 5;
            const int h  = h0 + i;
            const int w  = w0 + c;
            float val = 0.0f;
            if (h >= 0 && h < 128 && w >= 0 && w < 128)
                val = x[(((size_t)(n * CIN + ci0 + ch) * 128) + h) * 128 + w];
            smem[(i * XS_COLS + c) * XS_STRIDE + ch] = val;
        }
        __syncthreads();

        // ---- phase 2: V = BT * d * BT^T per (tile, ch) -> V LDS ----
        for (int pp = tid; pp < TQ * CHUNK; pp += 256) {
            const int tile = pp >> 5;
            const int ch   = pp & 31;
            float d[6][6];
#pragma unroll
            for (int i = 0; i < 6; ++i)
#pragma unroll
                for (int j = 0; j < 6; ++j)
                    d[i][j] = smem[(i * XS_COLS + tile * 4 + j) * XS_STRIDE + ch];

            float u[6][6];
#pragma unroll
            for (int j = 0; j < 6; ++j) {
                u[0][j] = 4.f * d[0][j] - 5.f * d[2][j] + d[4][j];
                u[1][j] = -4.f * d[1][j] - 4.f * d[2][j] + d[3][j] + d[4][j];
                u[2][j] =  4.f * d[1][j] - 4.f * d[2][j] - d[3][j] + d[4][j];
                u[3][j] = -2.f * d[1][j] - d[2][j] + 2.f * d[3][j] + d[4][j];
                u[4][j] =  2.f * d[1][j] - d[2][j] - 2.f * d[3][j] + d[4][j];
                u[5][j] =  4.f * d[1][j] - 5.f * d[3][j] + d[5][j];
            }
#pragma unroll
            for (int a = 0; a < 6; ++a) {
                float v0 = 4.f * u[a][0] - 5.f * u[a][2] + u[a][4];
                float v1 = -4.f * u[a][1] - 4.f * u[a][2] + u[a][3] + u[a][4];
                float v2 =  4.f * u[a][1] - 4.f * u[a][2] - u[a][3] + u[a][4];
                float v3 = -2.f * u[a][1] - u[a][2] + 2.f * u[a][3] + u[a][4];
                float v4 =  2.f * u[a][1] - u[a][2] - 2.f * u[a][3] + u[a][4];
                float v5 =  4.f * u[a][1] - 5.f * u[a][3] + u[a][5];
                float* vp = &smem[VOFF + (a * 6) * V_ABSTRIDE + tile * V_TSTRIDE + ch];
                vp[0 * V_ABSTRIDE] = v0;
                vp[1 * V_ABSTRIDE] = v1;
                vp[2 * V_ABSTRIDE] = v2;
                vp[3 * V_ABSTRIDE] = v3;
                vp[4 * V_ABSTRIDE] = v4;
                vp[5 * V_ABSTRIDE] = v5;
            }
        }
        __syncthreads();

        // ---- phase 3: per-wave WMMA GEMM, K = this Cin chunk (32) ----
        // A: 16 pos x 4 ci  (lanes 0-15 rows; VGPR0/1 = K0/K1, halves split K)
        // B: 4 ci x 16 co   (mirrored layout), from repacked weights in L2
#pragma unroll
        for (int j = 0; j < 5; ++j) {
            const int ab = wv + j * 8;
            if (ab < 36) {
                const float* wp0 = wt + ((size_t)ab * CIN + ci0) * COUT;
                const float* vp0 = &smem[VOFF + ab * V_ABSTRIDE + l16 * V_TSTRIDE];
#pragma unroll
                for (int k = 0; k < 8; ++k) {
                    const int kk = k * 4 + half * 2;
                    v2f A;
                    A.x = vp0[kk];
                    A.y = vp0[kk + 1];
#pragma unroll
                    for (int ct = 0; ct < 4; ++ct) {
                        const float* wp = wp0 + kk * COUT + ct * 16 + l16;
                        v2f B;
                        B.x = wp[0];
                        B.y = wp[COUT];
                        acc[j][ct] = __builtin_amdgcn_wmma_f32_16x16x4_f32(
                            false, A, false, B, (short)0, acc[j][ct], false, false);
                    }
                }
            }
        }
        __syncthreads();   // all V readers done before next chunk's transform
    }

    // ---- dump M accumulators to LDS (aliases xs/V region) ----
#pragma unroll
    for (int j = 0; j < 5; ++j) {
        const int ab = wv + j * 8;
        if (ab < 36) {
#pragma unroll
            for (int ct = 0; ct < 4; ++ct)
#pragma unroll
                for (int v = 0; v < 8; ++v) {
                    const int row = v + half * 8;
                    smem[(ab * 16 + row) * M_COSTRIDE + ct * 16 + l16] = acc[j][ct][v];
                }
        }
    }
    __syncthreads();

    // ---- phase 4: Y = AT * M * AT^T + bias, float4 stores (lanes over tiles) ----
    for (int r = 0; r < 4; ++r) {
        const int pp   = tid + 256 * r;      // 0..1023
        const int tile = pp & 15;
        const int co   = pp >> 4;

        float m[6][6];
#pragma unroll
        for (int a = 0; a < 6; ++a)
#pragma unroll
            for (int b = 0; b < 6; ++b)
                m[a][b] = smem[((a * 6 + b) * 16 + tile) * M_COSTRIDE + co];

        float t[4][6];
#pragma unroll
        for (int b = 0; b < 6; ++b) {
            t[0][b] = m[0][b] + m[1][b] + m[2][b] + m[3][b] + m[4][b];
            t[1][b] = m[1][b] - m[2][b] + 2.f * m[3][b] - 2.f * m[4][b];
            t[2][b] = m[1][b] + m[2][b] + 4.f * m[3][b] + 4.f * m[4][b];
            t[3][b] = m[1][b] - m[2][b] + 8.f * m[3][b] - 8.f * m[4][b] + m[5][b];
        }

        const float bv = bias[co];
        const int ow = (qStart + tile) * 4;
#pragma unroll
        for (int xr = 0; xr < 4; ++xr) {
            float4 out;
            out.x = t[xr][0] + t[xr][1] + t[xr][2] + t[xr][3] + t[xr][4] + bv;
            out.y = t[xr][1] - t[xr][2] + 2.f * t[xr][3] - 2.f * t[xr][4] + bv;
            out.z = t[xr][1] + t[xr][2] + 4.f * t[xr][3] + 4.f * t[xr][4] + bv;
            out.w = t[xr][1] - t[xr][2] + 8.f * t[xr][3] - 8.f * t[xr][4] + t[xr][5] + bv;
            const int oh = p * 4 + xr;
            *(float4*)&y[(((size_t)(n * COUT + co) * 128) + oh) * 128 + ow] = out;
        }
    }
}

// ---------------------------------------------------------------------------
extern "C" void kernel_launch(void* const* d_in, const int* in_sizes, int n_in,
                              void* d_out, int out_size, void* d_ws, size_t ws_size,
                              hipStream_t stream) {
    const float* x = (const float*)d_in[0];
    const float* w = (const float*)d_in[1];
    const float* b = (const float*)d_in[2];
    float* y  = (float*)d_out;
    float* wt = (float*)d_ws;     // 36*64*64 floats = 576 KB repacked weights

    repack_weights<<<(36 * CIN * COUT + 255) / 256, 256, 0, stream>>>(w, wt);

    dim3 grid(2, 32, 16);         // (q-blocks of 16 tiles, p tile-rows, N)
    winograd_fused<<<grid, 256, DYN_LDS_BYTES, stream>>>(x, wt, b, y);
}